// BaseViewTransform_60129542941
// MI455X (gfx1250) — compile-verified
//
#include <hip/hip_runtime.h>

#ifndef __has_builtin
#define __has_builtin(x) 0
#endif

// gfx1250 async-to-LDS probe: fall back to plain b128 loads if the builtins
// are not exposed by this toolchain.
#if __has_builtin(__builtin_amdgcn_global_load_async_to_lds_b128) && \
    __has_builtin(__builtin_amdgcn_s_wait_asynccnt)
#define USE_ASYNC_LDS 1
#else
#define USE_ASYNC_LDS 0
#endif

#define NCAM 6
#define DEPTH 118
#define FH 32
#define FW 88
#define CCH 80
#define NXV 360
#define NYV 360
#define NVOX (NXV * NYV)                    // 129600
#define NPTS (NCAM * DEPTH * FH * FW)       // 1,993,728
#define OUTSZ (CCH * NVOX)                  // 10,368,000
#define CGRP (CCH / 4)                      // 20 float4 groups per point
#define SCATTER_WORK ((long long)NPTS * CGRP) // 39,874,560 (divisible by 256)

#if USE_ASYNC_LDS
// Builtin's parameter type per hipcc diagnostic: int __vector_size__(16) *
typedef int v4i __attribute__((vector_size(16)));
typedef v4i __attribute__((address_space(1)))* gbl_v4i_ptr;  // global
typedef v4i __attribute__((address_space(3)))* lds_v4i_ptr;  // LDS
#endif

// -------------------------------------------------------------------------
// Relaxed agent-scope fp32 atomic add (no return) -> global_atomic_add_f32
// -------------------------------------------------------------------------
__device__ __forceinline__ void atomAddF32(float* p, float v) {
  (void)__hip_atomic_fetch_add(p, v, __ATOMIC_RELAXED, __HIP_MEMORY_SCOPE_AGENT);
}

// -------------------------------------------------------------------------
// Per-camera matrix folding (double precision for the 3x3 inverses)
// params layout per camera (24 floats):
//   [0..8]  P  = inv(post_rots)
//   [9..11] t  = post_trans
//   [12..20] M = extra_rots * c2l_rots * inv(intrins)
//   [21..23] T = extra_rots * c2l_trans + extra_trans
// -------------------------------------------------------------------------
__device__ void inv3d(const double* A, double* I) {
  double det = A[0] * (A[4] * A[8] - A[5] * A[7]) -
               A[1] * (A[3] * A[8] - A[5] * A[6]) +
               A[2] * (A[3] * A[7] - A[4] * A[6]);
  double r = 1.0 / det;
  I[0] = (A[4] * A[8] - A[5] * A[7]) * r;
  I[1] = (A[2] * A[7] - A[1] * A[8]) * r;
  I[2] = (A[1] * A[5] - A[2] * A[4]) * r;
  I[3] = (A[5] * A[6] - A[3] * A[8]) * r;
  I[4] = (A[0] * A[8] - A[2] * A[6]) * r;
  I[5] = (A[2] * A[3] - A[0] * A[5]) * r;
  I[6] = (A[3] * A[7] - A[4] * A[6]) * r;
  I[7] = (A[1] * A[6] - A[0] * A[7]) * r;
  I[8] = (A[0] * A[4] - A[1] * A[3]) * r;
}

__device__ void mm3d(const double* A, const double* B, double* C) {
  for (int r = 0; r < 3; ++r)
    for (int c = 0; c < 3; ++c)
      C[r * 3 + c] = A[r * 3 + 0] * B[0 * 3 + c] +
                     A[r * 3 + 1] * B[1 * 3 + c] +
                     A[r * 3 + 2] * B[2 * 3 + c];
}

__global__ void lss_params(const float* __restrict__ intr,
                           const float* __restrict__ c2l,
                           const float* __restrict__ imgaug,
                           const float* __restrict__ lidaug,
                           float* __restrict__ params) {
  const int n = threadIdx.x;
  if (n >= NCAM) return;
  double K[9], PR[9], CR[9], ER[9], ct[3], pt[3], et[3];
  for (int r = 0; r < 3; ++r) {
    for (int c = 0; c < 3; ++c) {
      K[r * 3 + c]  = (double)intr[n * 16 + r * 4 + c];
      CR[r * 3 + c] = (double)c2l[n * 16 + r * 4 + c];
      PR[r * 3 + c] = (double)imgaug[n * 16 + r * 4 + c];
      ER[r * 3 + c] = (double)lidaug[r * 4 + c];
    }
    ct[r] = (double)c2l[n * 16 + r * 4 + 3];
    pt[r] = (double)imgaug[n * 16 + r * 4 + 3];
    et[r] = (double)lidaug[r * 4 + 3];
  }
  double Ki[9], Pi[9], CMB[9], M[9], T[3];
  inv3d(K, Ki);
  inv3d(PR, Pi);
  mm3d(CR, Ki, CMB);
  mm3d(ER, CMB, M);
  for (int r = 0; r < 3; ++r)
    T[r] = ER[r * 3 + 0] * ct[0] + ER[r * 3 + 1] * ct[1] + ER[r * 3 + 2] * ct[2] + et[r];

  float* o = params + n * 24;
  for (int i = 0; i < 9; ++i) o[i] = (float)Pi[i];
  for (int r = 0; r < 3; ++r) o[9 + r] = (float)pt[r];
  for (int i = 0; i < 9; ++i) o[12 + i] = (float)M[i];
  for (int r = 0; r < 3; ++r) o[21 + r] = (float)T[r];
}

// -------------------------------------------------------------------------
// Zero output accumulator + counts (harness poisons both)
// -------------------------------------------------------------------------
__global__ __launch_bounds__(256) void lss_zero(float* __restrict__ out,
                                                float* __restrict__ cnts) {
  const long long i = ((long long)blockIdx.x * 256 + threadIdx.x) * 4;
  const float4 z = make_float4(0.f, 0.f, 0.f, 0.f);
  if (i < OUTSZ) {
    *(float4*)(out + i) = z;
  } else {
    const long long j = i - OUTSZ;
    if (j < NVOX) *(float4*)(cnts + j) = z;
  }
}

// -------------------------------------------------------------------------
// Scatter: one thread per (point, 4-channel group)
// -------------------------------------------------------------------------
__global__ __launch_bounds__(256) void lss_scatter(const float* __restrict__ feats,
                                                   const float* __restrict__ params,
                                                   float* __restrict__ out,
                                                   float* __restrict__ cnts) {
  const long long g = (long long)blockIdx.x * 256 + threadIdx.x;
  if (g >= SCATTER_WORK) return;  // exact multiple of 256: never taken

  // gfx1250: pull the next block's feature chunk toward L2 (global_prefetch_b8)
  __builtin_prefetch(feats + (g + 256) * 4, 0, 0);

#if USE_ASYNC_LDS
  __shared__ float4 stage[256];
  __builtin_amdgcn_global_load_async_to_lds_b128(
      (gbl_v4i_ptr)(feats + g * 4),
      (lds_v4i_ptr)&stage[threadIdx.x], 0, 0);
#else
  const float4 f = *(const float4*)(feats + g * 4);
#endif

  // ---- decompose work id (overlaps with the in-flight load) ----
  const int p  = (int)(g / CGRP);
  const int cg = (int)(g - (long long)p * CGRP);
  const int w  = p % FW;
  int t = p / FW;
  const int h = t % FH;
  t /= FH;
  const int dep = t % DEPTH;
  const int cam = t / DEPTH;

  const float* __restrict__ cp = params + cam * 24;

  // ---- frustum point (linspace over image coords, arange over depth) ----
  const float fx = (float)w * (703.0f / 87.0f);   // linspace(0, iW-1, fW)
  const float fy = (float)h * (255.0f / 31.0f);   // linspace(0, iH-1, fH)
  const float fz = 1.0f + 0.5f * (float)dep;      // arange(1.0, 60.0, 0.5)

  // ---- undo image aug: v = inv(post_rots) * (f - post_trans) ----
  const float ux = fx - cp[9], uy = fy - cp[10], uz = fz - cp[11];
  const float vx = cp[0] * ux + cp[1] * uy + cp[2] * uz;
  const float vy = cp[3] * ux + cp[4] * uy + cp[5] * uz;
  const float vz = cp[6] * ux + cp[7] * uy + cp[8] * uz;

  // ---- unproject (xy *= z), then fused cam->lidar->lidar-aug affine ----
  const float wx = vx * vz, wy = vy * vz, wz = vz;
  const float px = cp[12] * wx + cp[13] * wy + cp[14] * wz + cp[21];
  const float py = cp[15] * wx + cp[16] * wy + cp[17] * wz + cp[22];
  const float pz = cp[18] * wx + cp[19] * wy + cp[20] * wz + cp[23];

  // ---- quantize: ((pts - (BX - DX/2)) / DX).astype(int32)  (truncation!) ----
  const float lox = (-54.0f + 0.15f) - 0.15f;   // fold exactly as reference f32 math
  const int gx = (int)((px - lox) / 0.3f);
  const int gy = (int)((py - lox) / 0.3f);
  const int gz = (int)((pz + 10.0f) / 20.0f);

  const bool kept = (gx >= 0) & (gx < NXV) & (gy >= 0) & (gy < NYV) & (gz == 0);

#if USE_ASYNC_LDS
  __builtin_amdgcn_s_wait_asynccnt(0);
  const float4 f = stage[threadIdx.x];
#endif

  if (kept) {
    const int vox = gx * NYV + gy;
    float* base = out + (long long)(4 * cg) * NVOX + vox;
    atomAddF32(base,              f.x);
    atomAddF32(base + 1 * NVOX,   f.y);
    atomAddF32(base + 2 * NVOX,   f.z);
    atomAddF32(base + 3 * NVOX,   f.w);
    if (cg == 0) atomAddF32(cnts + vox, 1.0f);
  }
}

// -------------------------------------------------------------------------
// Normalize: out[c, x, y] /= max(count[x, y], 1)
// -------------------------------------------------------------------------
__global__ __launch_bounds__(256) void lss_normalize(float* __restrict__ out,
                                                     const float* __restrict__ cnts) {
  const long long i = ((long long)blockIdx.x * 256 + threadIdx.x) * 4;
  if (i >= OUTSZ) return;
  const int v = (int)(i % NVOX);  // NVOX divisible by 4 -> v is 16B aligned
  const float4 c = *(const float4*)(cnts + v);
  float4 o = *(float4*)(out + i);
  o.x = o.x / fmaxf(c.x, 1.0f);
  o.y = o.y / fmaxf(c.y, 1.0f);
  o.z = o.z / fmaxf(c.z, 1.0f);
  o.w = o.w / fmaxf(c.w, 1.0f);
  *(float4*)(out + i) = o;
}

// -------------------------------------------------------------------------
// Host entry
// -------------------------------------------------------------------------
extern "C" void kernel_launch(void* const* d_in, const int* in_sizes, int n_in,
                              void* d_out, int out_size, void* d_ws, size_t ws_size,
                              hipStream_t stream) {
  (void)in_sizes; (void)n_in; (void)out_size; (void)ws_size;

  const float* feats  = (const float*)d_in[0];  // (1,6,118,32,88,80) f32
  const float* intr   = (const float*)d_in[1];  // (1,6,4,4)
  const float* c2l    = (const float*)d_in[2];  // (1,6,4,4)
  const float* imgaug = (const float*)d_in[3];  // (1,6,4,4)
  const float* lidaug = (const float*)d_in[4];  // (1,4,4)

  float* out    = (float*)d_out;                       // (80,360,360) f32
  float* params = (float*)d_ws;                        // 144 floats
  float* cnts   = (float*)((char*)d_ws + 4096);        // 129600 floats

  lss_params<<<1, 32, 0, stream>>>(intr, c2l, imgaug, lidaug, params);

  const int zeroBlocks = (int)((OUTSZ + NVOX + 1023) / 1024);          // 10252
  lss_zero<<<zeroBlocks, 256, 0, stream>>>(out, cnts);

  const int scatBlocks = (int)(SCATTER_WORK / 256);                    // 155760
  lss_scatter<<<scatBlocks, 256, 0, stream>>>(feats, params, out, cnts);

  const int normBlocks = (int)(OUTSZ / 1024);                          // 10125
  lss_normalize<<<normBlocks, 256, 0, stream>>>(out, cnts);
}